// GCNConvThr_33191507263709
// MI455X (gfx1250) — compile-verified
//
#include <hip/hip_runtime.h>
#include <hip/hip_bf16.h>
#include <math.h>

#define F 256            // F_IN == F_OUT == 256 in the reference
#define THR_A_C 0.1f
#define THR_W_C 0.1f

typedef __attribute__((ext_vector_type(2))) float v2f;
typedef __attribute__((ext_vector_type(8))) float v8f;

// ---------------------------------------------------------------- zero scratch
__global__ void zero_kernel(float* colsq, float* sum_msg, int* locked, int n) {
    int i = blockIdx.x * blockDim.x + threadIdx.x;
    if (i < F) colsq[i] = 0.0f;
    if (i == F) *sum_msg = 0.0f;
    for (int j = i; j < n; j += gridDim.x * blockDim.x) locked[j] = 0;
}

// ------------------------------------------------- column sum-of-squares of x
// block = 256 threads, thread f owns column f; blocks stride over rows (coalesced)
__global__ void colsq_kernel(const float* __restrict__ x, float* __restrict__ colsq, int n) {
    int f = threadIdx.x;
    float acc = 0.0f;
    for (int r = blockIdx.x; r < n; r += gridDim.x) {
        float v = x[(size_t)r * F + f];
        acc += v * v;
    }
    atomicAdd(&colsq[f], acc);
}

// ---------------------------------------- per-column thresholds + prune flag
__global__ void thr_kernel(const float* __restrict__ colsq, float* __restrict__ thr,
                           float* __restrict__ flagp) {
    __shared__ float s[F];
    int f = threadIdx.x;
    float v = colsq[f];
    s[f] = v;
    __syncthreads();
    for (int off = F / 2; off > 0; off >>= 1) {
        if (f < off) s[f] += s[f + off];
        __syncthreads();
    }
    // ||norm_node_in|| = sqrt(sum of all x^2); norm_all_in = that / F
    float norm_all = sqrtf(s[0]) / (float)F;
    float norm_node = sqrtf(v);
    thr[f] = THR_W_C * norm_all / norm_node;
    if (f == 0) *flagp = (norm_all > 1e-8f) ? 1.0f : 0.0f;
}

// ------------------------------------------------------------------- prune W
__global__ void prune_kernel(const float* __restrict__ W, const float* __restrict__ thr,
                             const float* __restrict__ flagp, float* __restrict__ Weff) {
    int i = blockIdx.x * blockDim.x + threadIdx.x;   // F*F total
    int j = i & (F - 1);                             // input-column index
    float w = W[i];
    bool prune = (*flagp > 0.5f) && (fabsf(w) < thr[j]);
    Weff[i] = prune ? 0.0f : w;
}

// ------------------------------------------------ h = x @ Weff^T  (fp32 WMMA)
// One wave -> one 16x16 tile of h. K stepped by 4 via V_WMMA_F32_16X16X4_F32.
// A (16x4) lane layout: lanes 0-15 hold K={0,1} in v[0],v[1]; lanes 16-31 K={2,3}.
// B (4x16) mirrored: lane n = lane&15, same K split. B[k][n] = Weff[n][k].
__global__ void gemm_wmma_kernel(const float* __restrict__ x, const float* __restrict__ w,
                                 float* __restrict__ h, int n) {
    int wave = threadIdx.x >> 5;
    int lane = threadIdx.x & 31;
    int row0 = blockIdx.x * 16;                 // M tile
    int col0 = (blockIdx.y * 4 + wave) * 16;    // N tile (grid.y=4, 4 waves -> 16 tiles)

    int ml    = lane & 15;
    int khalf = lane >> 4;                      // 0: K={0,1}, 1: K={2,3}

    int arow = row0 + ml; if (arow >= n) arow = n - 1;      // clamp (dup loads, stores guarded)
    const float* xrow = x + (size_t)arow * F;
    const float* wrow = w + (size_t)(col0 + ml) * F;

    v8f acc = {};
    #pragma unroll 4
    for (int k = 0; k < F; k += 4) {
        v2f a, b;
        a.x = xrow[k + 2 * khalf];
        a.y = xrow[k + 2 * khalf + 1];
        b.x = wrow[k + 2 * khalf];
        b.y = wrow[k + 2 * khalf + 1];
        acc = __builtin_amdgcn_wmma_f32_16x16x4_f32(
            /*neg_a=*/false, a, /*neg_b=*/false, b,
            /*c_mod=*/(short)0, acc, /*reuse_a=*/false, /*reuse_b=*/false);
    }
    // C/D layout: VGPR r -> M = r (lanes 0-15) or r+8 (lanes 16-31); N = lane&15
    int nOut  = col0 + (lane & 15);
    int mBase = row0 + khalf * 8;
    float* hp = h + (size_t)mBase * F + nOut;
    if (row0 + 16 <= n) {
        // block-uniform fast path: full tile in range, branch-free stores
        #pragma unroll
        for (int r = 0; r < 8; ++r) hp[(size_t)r * F] = acc[r];
    } else {
        #pragma unroll
        for (int r = 0; r < 8; ++r)
            if (mBase + r < n) hp[(size_t)r * F] = acc[r];
    }
}

// ----------------------------------------------------------- per-row norms of h
__global__ void rownorm_kernel(const float* __restrict__ h, float* __restrict__ hnorm, int n) {
    int wave = threadIdx.x >> 5, lane = threadIdx.x & 31;
    int row = blockIdx.x * 8 + wave;
    if (row >= n) return;
    const float* hr = h + (size_t)row * F;
    float acc = 0.0f;
    #pragma unroll
    for (int j = lane; j < F; j += 32) { float v = hr[j]; acc += v * v; }
    #pragma unroll
    for (int off = 16; off > 0; off >>= 1) acc += __shfl_xor(acc, off, 32);
    if (lane == 0) hnorm[row] = sqrtf(acc);
}

// --------------------------------- sum over edges of ||msg_e|| = |w_e|*||h[src]||
__global__ void edgesum_kernel(const int* __restrict__ ei, const float* __restrict__ ew,
                               const float* __restrict__ hnorm, float* __restrict__ sum_msg,
                               int m) {
    __shared__ float s[256];
    float acc = 0.0f;
    for (int e = blockIdx.x * blockDim.x + threadIdx.x; e < m; e += gridDim.x * blockDim.x)
        acc += fabsf(ew[e]) * hnorm[ei[e]];
    s[threadIdx.x] = acc;
    __syncthreads();
    for (int off = 128; off > 0; off >>= 1) {
        if (threadIdx.x < off) s[threadIdx.x] += s[threadIdx.x + off];
        __syncthreads();
    }
    if (threadIdx.x == 0) atomicAdd(sum_msg, s[0]);
}

// ------------------------------------------------------------------ locked nodes
__global__ void lock_kernel(const int* __restrict__ nl, int* __restrict__ locked, int nlock) {
    int i = blockIdx.x * blockDim.x + threadIdx.x;
    if (i < nlock) locked[nl[i]] = 1;
}

// ------------------------------------------------------------- out = bias bcast
__global__ void initout_kernel(float* __restrict__ out, const float* __restrict__ b, int n) {
    size_t total = (size_t)n * F;
    for (size_t i = blockIdx.x * (size_t)blockDim.x + threadIdx.x; i < total;
         i += (size_t)gridDim.x * blockDim.x)
        out[i] = b[i & (F - 1)];
}

// ----------------------------- masked scatter-add: one wave per edge, 8 f/lane
__global__ void scatter_kernel(const int* __restrict__ ei, const float* __restrict__ ew,
                               const float* __restrict__ h, const float* __restrict__ hnorm,
                               const float* __restrict__ sum_msg, const int* __restrict__ locked,
                               float* __restrict__ out, int m) {
    int wave = threadIdx.x >> 5, lane = threadIdx.x & 31;
    int e = blockIdx.x * 8 + wave;
    if (e >= m) return;
    int src = ei[e];
    int dst = ei[m + e];
    float w = ew[e];
    float nf  = fabsf(w) * hnorm[src];
    float thr = THR_A_C * (*sum_msg) / (float)m;
    bool lock = (locked[dst] != 0) || (src == dst);
    if (nf < thr && !lock) return;   // wave-uniform: pruned edge contributes zero
    const float* hs = h + (size_t)src * F;
    float* od = out + (size_t)dst * F;
    #pragma unroll
    for (int j = 0; j < 8; ++j) {
        int f = lane + 32 * j;      // consecutive addresses across lanes per atomic
        atomicAdd(&od[f], w * hs[f]);
    }
}

extern "C" void kernel_launch(void* const* d_in, const int* in_sizes, int n_in,
                              void* d_out, int out_size, void* d_ws, size_t ws_size,
                              hipStream_t stream) {
    const float* x  = (const float*)d_in[0];
    const int*   ei = (const int*)d_in[1];
    const float* ew = (const float*)d_in[2];
    const int*   nl = (const int*)d_in[3];
    const float* W  = (const float*)d_in[4];
    const float* b  = (const float*)d_in[5];
    float* out = (float*)d_out;

    int n     = in_sizes[0] / F;     // 50000
    int m     = in_sizes[1] / 2;     // 400000
    int nlock = in_sizes[3];         // 1000

    // scratch carve-out
    char* ws = (char*)d_ws;
    size_t off = 0;
    auto alloc = [&](size_t bytes) -> void* {
        off = (off + 255) & ~(size_t)255;
        void* p = ws + off;
        off += bytes;
        return p;
    };
    float* h      = (float*)alloc((size_t)n * F * sizeof(float));   // 51.2 MB
    float* colsq  = (float*)alloc(F * sizeof(float));
    float* thr    = (float*)alloc(F * sizeof(float));
    float* flag   = (float*)alloc(sizeof(float));
    float* Weff   = (float*)alloc((size_t)F * F * sizeof(float));
    float* hnorm  = (float*)alloc((size_t)n * sizeof(float));
    float* sum_msg= (float*)alloc(sizeof(float));
    int*   locked = (int*)alloc((size_t)n * sizeof(int));
    (void)ws_size; (void)n_in; (void)out_size;

    zero_kernel<<<512, 256, 0, stream>>>(colsq, sum_msg, locked, n);
    colsq_kernel<<<256, 256, 0, stream>>>(x, colsq, n);
    thr_kernel<<<1, F, 0, stream>>>(colsq, thr, flag);
    prune_kernel<<<F, F, 0, stream>>>(W, thr, flag, Weff);

    dim3 ggrid((n + 15) / 16, 4);    // 3125 x 4 blocks, 4 waves/block -> 16 N-tiles
    gemm_wmma_kernel<<<ggrid, 128, 0, stream>>>(x, Weff, h, n);

    rownorm_kernel<<<(n + 7) / 8, 256, 0, stream>>>(h, hnorm, n);
    edgesum_kernel<<<1024, 256, 0, stream>>>(ei, ew, hnorm, sum_msg, m);
    lock_kernel<<<(nlock + 255) / 256, 256, 0, stream>>>(nl, locked, nlock);
    initout_kernel<<<2048, 256, 0, stream>>>(out, b, n);
    scatter_kernel<<<(m + 7) / 8, 256, 0, stream>>>(ei, ew, h, hnorm, sum_msg, locked, out, m);
}